// HeCoGCNConv_38439957299970
// MI455X (gfx1250) — compile-verified
//
#include <hip/hip_runtime.h>
#include <hip/hip_bf16.h>

typedef __attribute__((ext_vector_type(2))) float v2f;
typedef __attribute__((ext_vector_type(8))) float v8f;

#define N_NODES 100000
#define N_EDGES 1600000
#define DIM     128

// ---------------------------------------------------------------------------
// 1) degree with self loops: deg[i] = 1 + #incoming edges
// ---------------------------------------------------------------------------
__global__ void init_deg_kernel(float* __restrict__ deg, int n) {
    int i = blockIdx.x * blockDim.x + threadIdx.x;
    if (i < n) deg[i] = 1.0f;   // self loop contributes 1
}

__global__ void accum_deg_kernel(const int* __restrict__ dst, float* __restrict__ deg, int e) {
    int i = blockIdx.x * blockDim.x + threadIdx.x;
    if (i < e) atomicAdd(&deg[dst[i]], 1.0f);
}

__global__ void make_dinv_kernel(float* __restrict__ dinv, int n) {
    int i = blockIdx.x * blockDim.x + threadIdx.x;
    if (i < n) {
        float d = dinv[i];              // deg >= 1 always (self loop)
        dinv[i] = rsqrtf(d);
    }
}

// ---------------------------------------------------------------------------
// 2) h = x @ W.T  via V_WMMA_F32_16X16X4_F32
//    Block: 256 threads = 8 waves; block b owns rows [16b, 16b+16);
//    wave w owns output cols [16w, 16w+16).  K loop: 128 / 4 = 32 WMMAs.
//
//    ISA layouts (wave32):
//      A 16x4 f32 : lane l(0..15) v0=K0,v1=K1 ; lane 16..31 v0=K2,v1=K3
//                   -> lane loads float2 x[m0+ (l&15)][k + 2*(l>>4)]
//      B 4x16 f32 : same K striping across lane halves, N across lanes
//                   -> lane loads float2 W[n0 + (l&15)][k + 2*(l>>4)]  (B = W.T)
//      C/D 16x16  : vgpr v, lanes0-15 -> M=v, lanes16-31 -> M=v+8, N=lane&15
// ---------------------------------------------------------------------------
__global__ void __launch_bounds__(256)
gemm_xWt_kernel(const float* __restrict__ x, const float* __restrict__ W,
                float* __restrict__ h) {
    const int wave = threadIdx.x >> 5;          // 0..7  -> n-tile
    const int lane = threadIdx.x & 31;
    const int half = lane >> 4;                 // 0/1 -> K sub-pair
    const int l    = lane & 15;

    const int m0 = blockIdx.x * 16;             // N_NODES % 16 == 0
    const int n0 = wave * 16;

    const float* __restrict__ xrow = x + (size_t)(m0 + l) * DIM + 2 * half;
    const float* __restrict__ wrow = W + (size_t)(n0 + l) * DIM + 2 * half;

    v8f acc = {};
#pragma unroll
    for (int k = 0; k < DIM; k += 4) {
        v2f a = *(const v2f*)(xrow + k);
        v2f b = *(const v2f*)(wrow + k);
        // (neg_a, A, neg_b, B, c_mod, C, reuse_a, reuse_b)
        acc = __builtin_amdgcn_wmma_f32_16x16x4_f32(
            false, a, false, b, (short)0, acc, false, false);
    }

    float* __restrict__ hb = h + (size_t)(m0 + 8 * half) * DIM + n0 + l;
#pragma unroll
    for (int v = 0; v < 8; ++v) hb[(size_t)v * DIM] = acc[v];
}

// ---------------------------------------------------------------------------
// 3) out = b + dinv[i]^2 * h[i]   (bias + self-loop message, re-inits d_out)
// ---------------------------------------------------------------------------
__global__ void init_out_kernel(const float* __restrict__ h,
                                const float* __restrict__ dinv,
                                const float* __restrict__ bias,
                                float* __restrict__ out, int total) {
    int idx = blockIdx.x * blockDim.x + threadIdx.x;
    if (idx < total) {
        int node = idx >> 7;       // /128
        int d    = idx & (DIM - 1);
        float di = dinv[node];
        out[idx] = bias[d] + di * di * h[idx];
    }
}

// ---------------------------------------------------------------------------
// 4) scatter: one wave per edge; lane handles 4 contiguous floats
// ---------------------------------------------------------------------------
__global__ void __launch_bounds__(256)
scatter_edges_kernel(const int* __restrict__ src, const int* __restrict__ dst,
                     const float* __restrict__ h, const float* __restrict__ dinv,
                     float* __restrict__ out, int e) {
    int wid  = (int)((blockIdx.x * (unsigned)blockDim.x + threadIdx.x) >> 5);
    int lane = threadIdx.x & 31;
    if (wid >= e) return;

    int s = src[wid];
    int t = dst[wid];
    float w = dinv[s] * dinv[t];

    const float4 hv = *(const float4*)(h + (size_t)s * DIM + lane * 4);
    float* ob = out + (size_t)t * DIM + lane * 4;
    atomicAdd(ob + 0, w * hv.x);
    atomicAdd(ob + 1, w * hv.y);
    atomicAdd(ob + 2, w * hv.z);
    atomicAdd(ob + 3, w * hv.w);
}

// ---------------------------------------------------------------------------
// 5) PReLU in place
// ---------------------------------------------------------------------------
__global__ void prelu_kernel(float* __restrict__ out, const float* __restrict__ a, int total) {
    int idx = blockIdx.x * blockDim.x + threadIdx.x;
    if (idx < total) {
        float v = out[idx];
        float slope = a[0];
        out[idx] = (v >= 0.0f) ? v : slope * v;
    }
}

// ---------------------------------------------------------------------------
extern "C" void kernel_launch(void* const* d_in, const int* in_sizes, int n_in,
                              void* d_out, int out_size, void* d_ws, size_t ws_size,
                              hipStream_t stream) {
    const float* x    = (const float*)d_in[0];          // [N, 128]
    const int*   ei   = (const int*)d_in[1];            // [2, E] flat
    const float* W    = (const float*)d_in[2];          // [128, 128]
    const float* bias = (const float*)d_in[3];          // [128]
    const float* a    = (const float*)d_in[4];          // [1]
    float*       out  = (float*)d_out;                  // [N, 128]

    const int n = N_NODES;
    const int e = N_EDGES;
    const int total = n * DIM;

    const int* src = ei;        // edge_index[0]
    const int* dst = ei + e;    // edge_index[1]

    // workspace layout: dinv [N floats] | h [N*128 floats]
    char* ws = (char*)d_ws;
    float* dinv = (float*)ws;
    size_t h_off = (((size_t)n * sizeof(float)) + 255) & ~(size_t)255;
    float* h = (float*)(ws + h_off);

    const int T = 256;

    // degrees -> dinv
    init_deg_kernel<<<(n + T - 1) / T, T, 0, stream>>>(dinv, n);
    accum_deg_kernel<<<(e + T - 1) / T, T, 0, stream>>>(dst, dinv, e);
    make_dinv_kernel<<<(n + T - 1) / T, T, 0, stream>>>(dinv, n);

    // h = x @ W.T  (WMMA f32): one block per 16-row tile, 8 waves cover 128 cols
    gemm_xWt_kernel<<<n / 16, 256, 0, stream>>>(x, W, h);

    // out = bias + self-loop term
    init_out_kernel<<<(total + T - 1) / T, T, 0, stream>>>(h, dinv, bias, out, total);

    // scatter edges: 8 edges (waves) per block
    scatter_edges_kernel<<<(e + 7) / 8, 256, 0, stream>>>(src, dst, h, dinv, out, e);

    // PReLU
    prelu_kernel<<<(total + T - 1) / T, T, 0, stream>>>(out, a, total);
}